// NNEncoder_76304388980920
// MI455X (gfx1250) — compile-verified
//
#include <hip/hip_runtime.h>

typedef __attribute__((ext_vector_type(2))) float v2f;
typedef __attribute__((ext_vector_type(4))) float v4f;
typedef __attribute__((ext_vector_type(8))) float v8f;

#define IN_FEATS  64
#define OUT_FEATS 100

// ---------------------------------------------------------------------------
// Kernel 1: scatter-add edge features into per-node sums + degree counters.
// 16 lanes per edge, each lane handles one 16B chunk (coalesced loads).
// e (410MB) is streamed once -> non-temporal loads, so the 25.6MB node_sum
// accumulator stays resident in the 192MB L2 where the atomics resolve.
// ---------------------------------------------------------------------------
__global__ void gnn_scatter_kernel(const float* __restrict__ e,
                                   const int* __restrict__ dst,
                                   float* __restrict__ node_sum,
                                   float* __restrict__ deg,
                                   int n_edges) {
    long tid = (long)blockIdx.x * blockDim.x + threadIdx.x;
    int edge = (int)(tid >> 4);
    int sub  = (int)(tid & 15);
    if (edge >= n_edges) return;

    int d = __builtin_nontemporal_load(dst + edge);
    const v4f* erow = (const v4f*)(e + (size_t)edge * IN_FEATS);
    v4f v = __builtin_nontemporal_load(erow + sub);

    float* outp = node_sum + (size_t)d * IN_FEATS + sub * 4;
    atomicAdd(outp + 0, v.x);
    atomicAdd(outp + 1, v.y);
    atomicAdd(outp + 2, v.z);
    atomicAdd(outp + 3, v.w);
    if (sub == 0) atomicAdd(deg + d, 1.0f);
}

// ---------------------------------------------------------------------------
// Kernel 2: h = (node_sum @ W^T) / deg + b   (0 where deg == 0)
// One wave per 16x16 output tile, K=64 via 16 x V_WMMA_F32_16X16X4_F32.
// The 1/deg row scaling commutes with the GEMM, so the inner loop is a pure
// load -> wmma chain; division happens once per output element in the
// epilogue (matches the reference's sums/max(deg,1) rounding).
//
// f32 16x16x4 operand layout (wave32):
//   A (16x4, MxK): lanes 0-15 -> M=lane,    v0 = K=4k+0, v1 = K=4k+1
//                  lanes 16-31-> M=lane-16, v0 = K=4k+2, v1 = K=4k+3
//   B (4x16, KxN): mirrored: lanes hold N=lane&15, same K split by half.
//   C/D (16x16):   vgpr i -> row M = i + 8*(lane>=16), col N = lane&15.
// ---------------------------------------------------------------------------
__global__ void gnn_gemm_wmma_kernel(const float* __restrict__ node_sum,
                                     const float* __restrict__ deg,
                                     const float* __restrict__ W,   // [OUT, IN]
                                     const float* __restrict__ b,
                                     float* __restrict__ h,         // [nodes, OUT]
                                     int n_nodes) {
    const int lane = threadIdx.x & 31;
    const int wave = threadIdx.x >> 5;
    const int waves_per_block = blockDim.x >> 5;

    const int ntiles_n = (OUT_FEATS + 15) / 16;            // 7
    long tile = (long)blockIdx.x * waves_per_block + wave; // wave-uniform
    int mtile = (int)(tile / ntiles_n);
    int ntile = (int)(tile % ntiles_n);
    if (mtile * 16 >= n_nodes) return;                     // wave-uniform exit

    const int m0 = mtile * 16;
    const int n0 = ntile * 16;
    const int half = lane >> 4;   // 0: lanes 0-15, 1: lanes 16-31
    const int l    = lane & 15;

    // A-row pointer for this lane (raw sums; mean applied in epilogue)
    const float* aptr = node_sum + (size_t)(m0 + l) * IN_FEATS + half * 2;

    // B-column (output feature) for this lane. Tail lanes (ncol >= 100) read
    // W row 0 (safe address); their garbage columns are never stored and
    // cannot contaminate valid columns (WMMA column N only reads lane N's B).
    const int   ncol   = n0 + l;
    const bool  nvalid = (ncol < OUT_FEATS);
    const float* bptr  = W + (size_t)(nvalid ? ncol : 0) * IN_FEATS + half * 2;

    v8f acc = {0.f, 0.f, 0.f, 0.f, 0.f, 0.f, 0.f, 0.f};
#pragma unroll
    for (int kk = 0; kk < IN_FEATS / 4; ++kk) {
        v2f av = *(const v2f*)(aptr + kk * 4);
        v2f bv = *(const v2f*)(bptr + kk * 4);
        acc = __builtin_amdgcn_wmma_f32_16x16x4_f32(
            /*neg_a=*/false, av, /*neg_b=*/false, bv,
            /*c_mod=*/(short)0, acc, /*reuse_a=*/false, /*reuse_b=*/false);
    }

    // Epilogue: mean (acc/deg) + bias, zero rows with deg==0, guarded store.
    const float bias = nvalid ? b[ncol] : 0.0f;
#pragma unroll
    for (int i = 0; i < 8; ++i) {
        const int row = m0 + i + half * 8;
        const float dgr = deg[row];
        const float val = (dgr > 0.0f) ? (acc[i] / dgr + bias) : 0.0f;
        if (nvalid) h[(size_t)row * OUT_FEATS + ncol] = val;
    }
}

// ---------------------------------------------------------------------------
extern "C" void kernel_launch(void* const* d_in, const int* in_sizes, int n_in,
                              void* d_out, int out_size, void* d_ws, size_t ws_size,
                              hipStream_t stream) {
    const float* e   = (const float*)d_in[0];
    const int*   dst = (const int*)d_in[1];
    const float* W   = (const float*)d_in[2];
    const float* b   = (const float*)d_in[3];

    const int n_edges = in_sizes[1];
    const int n_nodes = out_size / OUT_FEATS;

    float* node_sum = (float*)d_ws;
    float* deg      = node_sum + (size_t)n_nodes * IN_FEATS;

    // Zero the accumulators (capturable as a graph memset node)
    const size_t zbytes = ((size_t)n_nodes * IN_FEATS + (size_t)n_nodes) * sizeof(float);
    (void)hipMemsetAsync(d_ws, 0, zbytes, stream);

    // Scatter: 16 lanes per edge
    {
        const int threads = 256;
        const long total = (long)n_edges * 16;
        const int grid = (int)((total + threads - 1) / threads);
        gnn_scatter_kernel<<<grid, threads, 0, stream>>>(e, dst, node_sum, deg, n_edges);
    }

    // WMMA GEMM over nodes
    {
        const int mtiles = (n_nodes + 15) / 16;
        const int ntilesN = (OUT_FEATS + 15) / 16;
        const long tiles = (long)mtiles * ntilesN;
        const int waves_per_block = 4;                 // 128 threads
        const int grid = (int)((tiles + waves_per_block - 1) / waves_per_block);
        gnn_gemm_wmma_kernel<<<grid, waves_per_block * 32, 0, stream>>>(
            node_sum, deg, W, b, (float*)d_out, n_nodes);
    }
}